// SAGE_Mini_82849919140445
// MI455X (gfx1250) — compile-verified
//
#include <hip/hip_runtime.h>
#include <stdint.h>

typedef float v2f __attribute__((ext_vector_type(2)));
typedef float v8f __attribute__((ext_vector_type(8)));
typedef unsigned int v4u __attribute__((ext_vector_type(4)));
typedef int v8i __attribute__((ext_vector_type(8)));
typedef int v4i __attribute__((ext_vector_type(4)));

#define DIM 128  // feature width of aggregation inputs (x/h1/h2 are all 128 wide)

#if __has_builtin(__builtin_amdgcn_tensor_load_to_lds) && __has_builtin(__builtin_amdgcn_s_wait_tensorcnt)
#define HAVE_TDM 1
#else
#define HAVE_TDM 0
#endif

// ---------------------------------------------------------------------------
// helpers
// ---------------------------------------------------------------------------
__device__ __forceinline__ void atomic_add_f32(float* p, float v) {
    // relaxed, device-scope fp32 atomic add -> global_atomic_add_f32 (L2 atomics)
    __hip_atomic_fetch_add(p, v, __ATOMIC_RELAXED, __HIP_MEMORY_SCOPE_AGENT);
}

__global__ void zero_f32(float* __restrict__ p, int n) {
    int i = blockIdx.x * blockDim.x + threadIdx.x;
    if (i < n) p[i] = 0.0f;
}

__global__ void zero_f32x4(float4* __restrict__ p, int n4) {
    int i = blockIdx.x * blockDim.x + threadIdx.x;
    if (i < n4) p[i] = make_float4(0.f, 0.f, 0.f, 0.f);
}

// ---------------------------------------------------------------------------
// degree: deg[i] = #edges with dst == i, then rdeg[i] = 1/max(deg,1)
// ---------------------------------------------------------------------------
__global__ void degree_kernel(const int* __restrict__ dst, float* __restrict__ deg, int nedges) {
    int i = blockIdx.x * blockDim.x + threadIdx.x;
    if (i < nedges) atomic_add_f32(&deg[dst[i]], 1.0f);
}

__global__ void rdeg_kernel(float* __restrict__ deg, int n) {
    int i = blockIdx.x * blockDim.x + threadIdx.x;
    if (i < n) deg[i] = 1.0f / fmaxf(deg[i], 1.0f);
}

// ---------------------------------------------------------------------------
// scatter: agg[dst[e], :] += x[src[e], :]   (one wave32 per edge, float4/lane)
// ---------------------------------------------------------------------------
__global__ void scatter_kernel(const float* __restrict__ x,
                               const int* __restrict__ src,
                               const int* __restrict__ dst,
                               float* __restrict__ agg, int nedges) {
    int t = blockIdx.x * blockDim.x + threadIdx.x;
    int e = t >> 5;
    int lane = t & 31;
    if (e >= nedges) return;
    int s = src[e];
    int d = dst[e];
    const float4 v = *(const float4*)(x + (size_t)s * DIM + lane * 4);
    float* o = agg + (size_t)d * DIM + lane * 4;
    atomic_add_f32(o + 0, v.x);
    atomic_add_f32(o + 1, v.y);
    atomic_add_f32(o + 2, v.z);
    atomic_add_f32(o + 3, v.w);
}

// ---------------------------------------------------------------------------
// TDM: DMA a 2D fp32 tile (width x height, row stride `stride`, all in
// elements) from global memory into LDS at byte offset lds_off.
// D# layout per CDNA5 ISA ch.8 (group0 128b, group1 256b; groups 2/3 unused
// for 2D tensors). 6-arg builtin form (clang-23 / therock-10.0 headers).
// ---------------------------------------------------------------------------
#if HAVE_TDM
__device__ __forceinline__ void tdm_load_2d_f32(const void* gptr, uint32_t lds_off,
                                                int width, int height, int stride) {
    uint64_t ga = (uint64_t)(uintptr_t)gptr;
    v4u g0;
    g0[0] = 1u;                                                // count=1, user D#
    g0[1] = lds_off;                                           // lds_addr
    g0[2] = (uint32_t)(ga & 0xFFFFFFFFu);                      // global_addr[31:0]
    g0[3] = (uint32_t)((ga >> 32) & 0x01FFFFFFu) | 0x80000000u;// addr[56:32] | type=2
    v8i g1;
    g1[0] = (2 << 16);                                         // data_size=4B, mask=0
    g1[1] = (int)((width & 0xFFFF) << 16);                     // tensor_dim0[15:0]
    g1[2] = (int)(((width >> 16) & 0xFFFF) |                   // tensor_dim0[31:16]
                  ((height & 0xFFFF) << 16));                  // tensor_dim1[15:0]
    g1[3] = (int)(((height >> 16) & 0xFFFF) |                  // tensor_dim1[31:16]
                  ((width & 0xFFFF) << 16));                   // tile_dim0
    g1[4] = (int)(height & 0xFFFF);                            // tile_dim1 (tile_dim2=0)
    g1[5] = stride;                                            // tensor_dim0_stride[31:0]
    g1[6] = 0;
    g1[7] = 0;
    v4i z4 = {0, 0, 0, 0};
    v8i z8 = {0, 0, 0, 0, 0, 0, 0, 0};
    __builtin_amdgcn_tensor_load_to_lds(g0, g1, z4, z4, z8, 0);
}
#endif

// ---------------------------------------------------------------------------
// fused SAGE GEMM: out = relu?( (rdeg*agg) @ Wl + bl + xin @ Wr )
// - Wl/Wr staged in LDS (TDM if available) once per block
// - one wave owns a 16-row block and ALL DOUT/16 column tiles (A-frag reuse)
// - PASSES row-blocks per wave per block
// ---------------------------------------------------------------------------
template <int DOUT, bool RELU, int PASSES>
__global__ __launch_bounds__(128) void sage_gemm_kernel(
        const float* __restrict__ agg, const float* __restrict__ rdeg,
        const float* __restrict__ xin, const float* __restrict__ Wl,
        const float* __restrict__ bl, const float* __restrict__ Wr,
        float* __restrict__ out, int n) {
    constexpr int DI = 128;
    constexpr int NT = DOUT / 16;  // column tiles per wave
    __shared__ float lWl[DI * DOUT];
    __shared__ float lWr[DI * DOUT];

    const int lane = threadIdx.x & 31;
    const int wib  = threadIdx.x >> 5;  // wave in block (0..3)

    // ---- stage weights into LDS ----
#if HAVE_TDM
    if (wib == 0) {  // wave-uniform branch: only wave 0 issues the DMAs
        tdm_load_2d_f32(Wl, (uint32_t)(uintptr_t)(void*)lWl, DOUT, DI, DOUT);
        tdm_load_2d_f32(Wr, (uint32_t)(uintptr_t)(void*)lWr, DOUT, DI, DOUT);
        __builtin_amdgcn_s_wait_tensorcnt(0);
    }
#else
    {
        const int nelem = DI * DOUT;  // per matrix
        for (int i = threadIdx.x * 4; i < nelem; i += 128 * 4) {
            *(float4*)(lWl + i) = *(const float4*)(Wl + i);
            *(float4*)(lWr + i) = *(const float4*)(Wr + i);
        }
    }
#endif
    __syncthreads();

    const int nrb = (n + 15) >> 4;
    const int koff = (lane >> 4) << 1;   // 0 | 2 : A/B K-pair select per lane half
    const int ncol = lane & 15;          // B/C/D column within tile
    const int mofs = (lane >> 4) << 3;   // C layout: lanes 16-31 hold M=j+8

    for (int pass = 0; pass < PASSES; ++pass) {
        const int rb = blockIdx.x * (PASSES * 4) + pass * 4 + wib;
        if (rb >= nrb) continue;  // wave-uniform: EXEC stays all-ones for WMMA

        const int rowbase = rb * 16;
        int arow = rowbase + (lane & 15);
        if (arow >= n) arow = n - 1;  // clamp (no EXEC divergence)
        const float rd = rdeg[arow];
        const float* __restrict__ ar = agg + (size_t)arow * DI;
        const float* __restrict__ xr = xin + (size_t)arow * DI;

        v8f acc[NT];
#pragma unroll
        for (int t = 0; t < NT; ++t) acc[t] = (v8f){0.f, 0.f, 0.f, 0.f, 0.f, 0.f, 0.f, 0.f};

        // phase 1: (rdeg*agg) @ Wl (from LDS)
#pragma unroll 4
        for (int k = 0; k < DI; k += 4) {
            const float2 av = *(const float2*)(ar + k + koff);
            v2f a;
            a.x = rd * av.x;
            a.y = rd * av.y;
            const float* bp = lWl + (k + koff) * DOUT + ncol;
#pragma unroll
            for (int t = 0; t < NT; ++t) {
                v2f b;
                b.x = bp[t * 16];
                b.y = bp[DOUT + t * 16];
                acc[t] = __builtin_amdgcn_wmma_f32_16x16x4_f32(false, a, false, b, (short)0,
                                                               acc[t], false, false);
            }
        }
        // phase 2: xin @ Wr (from LDS)
#pragma unroll 4
        for (int k = 0; k < DI; k += 4) {
            const float2 av = *(const float2*)(xr + k + koff);
            v2f a;
            a.x = av.x;
            a.y = av.y;
            const float* bp = lWr + (k + koff) * DOUT + ncol;
#pragma unroll
            for (int t = 0; t < NT; ++t) {
                v2f b;
                b.x = bp[t * 16];
                b.y = bp[DOUT + t * 16];
                acc[t] = __builtin_amdgcn_wmma_f32_16x16x4_f32(false, a, false, b, (short)0,
                                                               acc[t], false, false);
            }
        }

        // store with bias (+ReLU)
#pragma unroll
        for (int t = 0; t < NT; ++t) {
            const int col = t * 16 + ncol;
            const float bias = bl[col];
#pragma unroll
            for (int j = 0; j < 8; ++j) {
                const int m = rowbase + mofs + j;
                float v = acc[t][j] + bias;
                if (RELU) v = fmaxf(v, 0.0f);
                if (m < n) out[(size_t)m * DOUT + col] = v;
            }
        }
    }
}

// ---------------------------------------------------------------------------
// launch
// ---------------------------------------------------------------------------
extern "C" void kernel_launch(void* const* d_in, const int* in_sizes, int n_in,
                              void* d_out, int out_size, void* d_ws, size_t ws_size,
                              hipStream_t stream) {
    const float* x   = (const float*)d_in[0];
    const int*   src = (const int*)d_in[1];
    const int*   dst = (const int*)d_in[2];
    const float* Wl0 = (const float*)d_in[3];
    const float* bl0 = (const float*)d_in[4];
    const float* Wr0 = (const float*)d_in[5];
    const float* Wl1 = (const float*)d_in[6];
    const float* bl1 = (const float*)d_in[7];
    const float* Wr1 = (const float*)d_in[8];
    const float* Wl2 = (const float*)d_in[9];
    const float* bl2 = (const float*)d_in[10];
    const float* Wr2 = (const float*)d_in[11];
    float* out = (float*)d_out;

    const int n = in_sizes[0] / DIM;  // 100000 nodes
    const int e = in_sizes[1];        // 1600000 edges

    char* ws = (char*)d_ws;
    size_t off = 0;
    float* rdeg = (float*)(ws + off);
    off = (off + (size_t)n * 4 + 255) & ~(size_t)255;
    float* agg = (float*)(ws + off);
    off = (off + (size_t)n * DIM * 4 + 255) & ~(size_t)255;
    float* h1 = (float*)(ws + off);
    off = (off + (size_t)n * DIM * 4 + 255) & ~(size_t)255;
    float* h2 = (float*)(ws + off);

    // degree -> reciprocal (shared by all layers)
    zero_f32<<<(n + 255) / 256, 256, 0, stream>>>(rdeg, n);
    degree_kernel<<<(e + 255) / 256, 256, 0, stream>>>(dst, rdeg, e);
    rdeg_kernel<<<(n + 255) / 256, 256, 0, stream>>>(rdeg, n);

    const int nagg4 = n * DIM / 4;
    const int scat_blocks = (e * 32 + 255) / 256;
    const int nrb = (n + 15) / 16;
    constexpr int PASSES = 4;
    const int gemm_blocks = (nrb + PASSES * 4 - 1) / (PASSES * 4);

    // ---- layer 0: x -> h1 (relu) ----
    zero_f32x4<<<(nagg4 + 255) / 256, 256, 0, stream>>>((float4*)agg, nagg4);
    scatter_kernel<<<scat_blocks, 256, 0, stream>>>(x, src, dst, agg, e);
    sage_gemm_kernel<128, true, PASSES>
        <<<gemm_blocks, 128, 0, stream>>>(agg, rdeg, x, Wl0, bl0, Wr0, h1, n);

    // ---- layer 1: h1 -> h2 (relu) ----
    zero_f32x4<<<(nagg4 + 255) / 256, 256, 0, stream>>>((float4*)agg, nagg4);
    scatter_kernel<<<scat_blocks, 256, 0, stream>>>(h1, src, dst, agg, e);
    sage_gemm_kernel<128, true, PASSES>
        <<<gemm_blocks, 128, 0, stream>>>(agg, rdeg, h1, Wl1, bl1, Wr1, h2, n);

    // ---- layer 2: h2 -> out (no relu) ----
    zero_f32x4<<<(nagg4 + 255) / 256, 256, 0, stream>>>((float4*)agg, nagg4);
    scatter_kernel<<<scat_blocks, 256, 0, stream>>>(h2, src, dst, agg, e);
    sage_gemm_kernel<64, false, PASSES>
        <<<gemm_blocks, 128, 0, stream>>>(agg, rdeg, h2, Wl2, bl2, Wr2, out, n);
}